// RWKV_RNN_34608846471185
// MI455X (gfx1250) — compile-verified
//
#include <hip/hip_runtime.h>
#include <hip/hip_bf16.h>
#include <math.h>

// ---------------- geometry ----------------
#define NL   24
#define D    1024
#define HID  4096
#define NWG  320
#define TPB  256
#define LNEPS 1e-5f

typedef __attribute__((ext_vector_type(2))) float v2f;
typedef __attribute__((ext_vector_type(8))) float v8f;

struct Params {
  const float* x;
  const float* state;
  const float* ln1_w; const float* ln1_b;
  const float* ln2_w; const float* ln2_b;
  const float* tmk; const float* tmv; const float* tmr;
  const float* tfirst; const float* tdecay;
  const float* att_kw; const float* att_vw; const float* att_rw; const float* att_ow;
  const float* ftmk; const float* ftmr;
  const float* ffn_kw; const float* ffn_vw; const float* ffn_rw;
  float* out;       // [D + NL*5*D]
  float* ws;        // float scratch region (after barrier counters)
  unsigned* bar;    // one counter per grid barrier
};

// ---------------- grid barrier (atomic counter, one slot per use) ----------
__device__ __forceinline__ void grid_barrier(unsigned* bar, int slot) {
  __threadfence();                 // drain this thread's global stores
  __syncthreads();
  if (threadIdx.x == 0) {
    __hip_atomic_fetch_add(&bar[slot], 1u, __ATOMIC_RELEASE, __HIP_MEMORY_SCOPE_AGENT);
    while (__hip_atomic_load(&bar[slot], __ATOMIC_ACQUIRE, __HIP_MEMORY_SCOPE_AGENT) < (unsigned)NWG) {
      __builtin_amdgcn_s_sleep(2);
    }
  }
  __syncthreads();
  __threadfence();                 // acquire: invalidate stale cached lines
}

// ---------------- block reduction ----------------
__device__ __forceinline__ float block_sum(float v, float* s_red) {
  const int tid = threadIdx.x;
  s_red[tid] = v;
  __syncthreads();
  for (int s = TPB / 2; s > 0; s >>= 1) {
    if (tid < s) s_red[tid] += s_red[tid + s];
    __syncthreads();
  }
  float r = s_red[0];
  __syncthreads();
  return r;
}

// ---------------- 16-row GEMV tile via V_WMMA_F32_16X16X4_F32 --------------
// W: row-major weight base (already offset to layer), ld: row stride (=K),
// s_x: LDS operand vector, K: reduction length (K % 32 == 0).
// K is split across the 8 waves; each wave accumulates a v8f partial and the
// lane-0 / lane-16 columns hold y[row0..row0+7] / y[row0+8..row0+15].
__device__ __forceinline__ void gemv16_partials(
    const float* __restrict__ W, int ld, int row0,
    const float* __restrict__ s_x, int K, float* s_part) {
  const int tid  = threadIdx.x;
  const int wave = tid >> 5;
  const int lane = tid & 31;           // wave32
  const int m    = lane & 15;          // output row within tile
  const int kq   = (lane >> 4) << 1;   // K sub-offset: 0 for lanes 0-15, 2 for 16-31
  const float* wr = W + (size_t)(row0 + m) * (size_t)ld + kq;
  const float* xr = s_x + kq;
  const int KW = K >> 3;               // K per wave
  const int k0 = wave * KW;
  v8f c = {};
  for (int kt = k0; kt < k0 + KW; kt += 4) {
    // A fragment: 16 rows x 4 k of the weight tile (streamed once -> NT)
    v2f a = __builtin_nontemporal_load((const v2f*)(wr + kt));
    // B fragment: x[kt..kt+3] replicated across the 16 N columns
    v2f b;
    b.x = xr[kt];
    b.y = xr[kt + 1];
    c = __builtin_amdgcn_wmma_f32_16x16x4_f32(
        /*neg_a=*/false, a, /*neg_b=*/false, b,
        /*c_mod=*/(short)0, c, /*reuse_a=*/false, /*reuse_b=*/false);
  }
  if (lane == 0) {
#pragma unroll
    for (int i = 0; i < 8; ++i) s_part[wave * 16 + i] = c[i];
  } else if (lane == 16) {
#pragma unroll
    for (int i = 0; i < 8; ++i) s_part[wave * 16 + 8 + i] = c[i];
  }
}

__device__ __forceinline__ float tile_reduce(const float* s_part, int r) {
  float y = 0.f;
#pragma unroll
  for (int w = 0; w < 8; ++w) y += s_part[w * 16 + r];
  return y;
}

__device__ __forceinline__ float sigmoidf_(float x) {
  return 1.0f / (1.0f + expf(-x));
}

// ---------------- persistent mega-kernel ----------------
__global__ void __launch_bounds__(TPB)
rwkv_forward(Params p) {
  __shared__ float sm[HID];        // 16KB: xn/xk/xv/xr (4x1024) or kk stage (4096)
  __shared__ float s_red[TPB];
  __shared__ float s_part[8 * 16];
  float* s_xn = sm;                // also xn2 in FFN phase
  float* s_xk = sm + 1024;         // also fxk
  float* s_xv = sm + 2048;
  float* s_xr = sm + 3072;         // also fxr

  const int wg  = blockIdx.x;
  const int tid = threadIdx.x;

  float* xc_buf   = p.ws;                 // D
  float* k_buf    = p.ws + 1024;          // D
  float* v_buf    = p.ws + 2048;          // D
  float* r_buf    = p.ws + 3072;          // D (already sigmoided)
  float* rwkv_buf = p.ws + 4096;          // D
  float* fr_buf   = p.ws + 5120;          // D (already sigmoided)
  float* kk_buf   = p.ws + 6144;          // H
  float* st_out   = p.out + D;            // state output region

  int bslot = 0;

  for (int l = 0; l < NL; ++l) {
    const size_t lD = (size_t)l * D;
    const float* xin = (l == 0) ? p.x : xc_buf;

    // ---- P1: LN1 + time-mix (every WG, redundant; feeds its own LDS) ----
    {
      float lsum = 0.f, lsq = 0.f;
      for (int i = tid; i < D; i += TPB) {
        float v = xin[i];
        s_xn[i] = v;
        lsum += v; lsq += v * v;
      }
      float tot  = block_sum(lsum, s_red);
      float tot2 = block_sum(lsq, s_red);
      float mean = tot * (1.0f / D);
      float rstd = rsqrtf(fmaxf(tot2 * (1.0f / D) - mean * mean, 0.0f) + LNEPS);
      const float* w1 = p.ln1_w + lD;
      const float* b1 = p.ln1_b + lD;
      const float* s1 = p.state + (lD * 5) + D;      // state row 1
      for (int i = tid; i < D; i += TPB) {
        float xnv = (s_xn[i] - mean) * rstd * w1[i] + b1[i];
        float sv  = s1[i];
        s_xn[i] = xnv;
        float a;
        a = p.tmk[lD + i]; s_xk[i] = xnv * a + sv * (1.0f - a);
        a = p.tmv[lD + i]; s_xv[i] = xnv * a + sv * (1.0f - a);
        a = p.tmr[lD + i]; s_xr[i] = xnv * a + sv * (1.0f - a);
      }
      __syncthreads();
    }

    // ---- P2: k = kw@xk, v = vw@xv, r = sigmoid(rw@xr) : 192 tiles ----
    for (int t = wg; t < 192; t += NWG) {
      const int mat  = t >> 6;
      const int row0 = (t & 63) << 4;
      const float* W =
          (mat == 0 ? p.att_kw : mat == 1 ? p.att_vw : p.att_rw) + lD * D;
      const float* sx = (mat == 0 ? s_xk : mat == 1 ? s_xv : s_xr);
      gemv16_partials(W, D, row0, sx, D, s_part);
      __syncthreads();
      if (tid < 16) {
        float y = tile_reduce(s_part, tid);
        int d = row0 + tid;
        if      (mat == 0) k_buf[d] = y;
        else if (mat == 1) v_buf[d] = y;
        else               r_buf[d] = sigmoidf_(y);
      }
      __syncthreads();
    }
    grid_barrier(p.bar, bslot++);

    // ---- P3: WKV elementwise + state rows 1..4 (WGs 0..3) ----
    if (wg < 4) {
      const int d = wg * TPB + tid;
      const size_t sb = lD * 5;
      float kk = k_buf[d], vv = v_buf[d], rr = r_buf[d];
      float aa = p.state[sb + 2 * D + d];
      float bb = p.state[sb + 3 * D + d];
      float pp = p.state[sb + 4 * D + d];
      float tf = p.tfirst[lD + d];
      float td = p.tdecay[lD + d];
      float ww = tf + kk;
      float pm = fmaxf(pp, ww);
      float e1 = expf(pp - pm), e2 = expf(ww - pm);
      float wkv = (e1 * aa + e2 * vv) / (e1 * bb + e2);
      float ww2 = pp + td;
      float p2  = fmaxf(ww2, kk);
      float f1 = expf(ww2 - p2), f2 = expf(kk - p2);
      st_out[sb + 1 * D + d] = s_xn[d];           // xn
      st_out[sb + 2 * D + d] = f1 * aa + f2 * vv; // aa_n
      st_out[sb + 3 * D + d] = f1 * bb + f2;      // bb_n
      st_out[sb + 4 * D + d] = p2;                // p2
      rwkv_buf[d] = rr * wkv;
    }
    grid_barrier(p.bar, bslot++);

    // ---- P4: xc += ow @ (r*wkv) : 64 tiles ----
    for (int t = wg; t < 64; t += NWG) {
      for (int i = tid; i < D; i += TPB) s_xn[i] = rwkv_buf[i];
      __syncthreads();
      const int row0 = t << 4;
      gemv16_partials(p.att_ow + lD * D, D, row0, s_xn, D, s_part);
      __syncthreads();
      if (tid < 16) {
        int d = row0 + tid;
        xc_buf[d] = xin[d] + tile_reduce(s_part, tid);
      }
      __syncthreads();
    }
    grid_barrier(p.bar, bslot++);

    // ---- P5: LN2 + channel-mix, then fr = sigmoid(frw@fxr) (64 tiles)
    //          and kk = relu(fkw@fxk)^2 (256 tiles) ----
    {
      float lsum = 0.f, lsq = 0.f;
      for (int i = tid; i < D; i += TPB) {
        float v = xc_buf[i];
        s_xn[i] = v;
        lsum += v; lsq += v * v;
      }
      float tot  = block_sum(lsum, s_red);
      float tot2 = block_sum(lsq, s_red);
      float mean = tot * (1.0f / D);
      float rstd = rsqrtf(fmaxf(tot2 * (1.0f / D) - mean * mean, 0.0f) + LNEPS);
      const float* w2 = p.ln2_w + lD;
      const float* b2 = p.ln2_b + lD;
      const float* s0 = p.state + (lD * 5);        // state row 0
      for (int i = tid; i < D; i += TPB) {
        float xnv = (s_xn[i] - mean) * rstd * w2[i] + b2[i];
        float sv  = s0[i];
        s_xn[i] = xnv;
        float a;
        a = p.ftmk[lD + i]; s_xk[i] = xnv * a + sv * (1.0f - a);
        a = p.ftmr[lD + i]; s_xr[i] = xnv * a + sv * (1.0f - a);
      }
      __syncthreads();
      if (wg == 0) {
        const size_t sb = lD * 5;
        for (int i = tid; i < D; i += TPB) st_out[sb + i] = s_xn[i]; // xn2
      }
    }
    for (int t = wg; t < 320; t += NWG) {
      if (t < 64) {
        const int row0 = t << 4;
        gemv16_partials(p.ffn_rw + lD * D, D, row0, s_xr, D, s_part);
        __syncthreads();
        if (tid < 16) fr_buf[row0 + tid] = sigmoidf_(tile_reduce(s_part, tid));
        __syncthreads();
      } else {
        const int row0 = (t - 64) << 4;
        gemv16_partials(p.ffn_kw + lD * HID, D, row0, s_xk, D, s_part);
        __syncthreads();
        if (tid < 16) {
          float y = fmaxf(tile_reduce(s_part, tid), 0.0f);
          kk_buf[row0 + tid] = y * y;
        }
        __syncthreads();
      }
    }
    grid_barrier(p.bar, bslot++);

    // ---- P6: xc = (xc + fr * (fvw @ kk)) * scale : 64 tiles, K = 4096 ----
    {
      const float sc = (((l + 1) % 6) == 0) ? 0.5f : 1.0f;
      for (int t = wg; t < 64; t += NWG) {
        for (int i = tid; i < HID; i += TPB) sm[i] = kk_buf[i];
        __syncthreads();
        const int row0 = t << 4;
        gemv16_partials(p.ffn_vw + lD * HID, HID, row0, sm, HID, s_part);
        __syncthreads();
        if (tid < 16) {
          int d = row0 + tid;
          xc_buf[d] = (xc_buf[d] + fr_buf[d] * tile_reduce(s_part, tid)) * sc;
        }
        __syncthreads();
      }
    }
    grid_barrier(p.bar, bslot++);
  }

  // ---- epilogue: x_out ----
  if (wg == 0) {
    for (int i = tid; i < D; i += TPB) p.out[i] = xc_buf[i];
  }
}

// ---------------- host entry ----------------
extern "C" void kernel_launch(void* const* d_in, const int* in_sizes, int n_in,
                              void* d_out, int out_size, void* d_ws, size_t ws_size,
                              hipStream_t stream) {
  (void)in_sizes; (void)n_in; (void)out_size; (void)ws_size;
  Params p;
  p.x       = (const float*)d_in[0];
  p.state   = (const float*)d_in[1];
  p.ln1_w   = (const float*)d_in[2];
  p.ln1_b   = (const float*)d_in[3];
  p.ln2_w   = (const float*)d_in[4];
  p.ln2_b   = (const float*)d_in[5];
  p.tmk     = (const float*)d_in[6];
  p.tmv     = (const float*)d_in[7];
  p.tmr     = (const float*)d_in[8];
  p.tfirst  = (const float*)d_in[9];
  p.tdecay  = (const float*)d_in[10];
  p.att_kw  = (const float*)d_in[11];
  p.att_vw  = (const float*)d_in[12];
  p.att_rw  = (const float*)d_in[13];
  p.att_ow  = (const float*)d_in[14];
  p.ftmk    = (const float*)d_in[15];
  p.ftmr    = (const float*)d_in[16];
  p.ffn_kw  = (const float*)d_in[17];
  p.ffn_vw  = (const float*)d_in[18];
  p.ffn_rw  = (const float*)d_in[19];
  p.out     = (float*)d_out;
  p.bar     = (unsigned*)d_ws;
  p.ws      = (float*)((char*)d_ws + 512);

  // reset the 120 grid-barrier counters (graph-capture-legal async memset)
  hipMemsetAsync(d_ws, 0, 512, stream);
  rwkv_forward<<<NWG, TPB, 0, stream>>>(p);
}